// GATConv_41377714929864
// MI455X (gfx1250) — compile-verified
//
#include <hip/hip_runtime.h>
#include <hip/hip_bf16.h>

#define GN 100000
#define GE 1600000
#define GDIN 128
#define GH 4
#define GD 32
#define GHD 128   // H*D
#define ROWS 32   // rows of h per block in the GEMM (100000 = 3125 * 32)
#define SWS 136   // padded LDS row stride for W (2*136 % 64 == 16 -> bank-clean)

typedef __attribute__((ext_vector_type(2))) float v2f;
typedef __attribute__((ext_vector_type(8))) float v8f;
typedef __attribute__((ext_vector_type(4))) int  v4i;

#define AS1 __attribute__((address_space(1)))
#define AS3 __attribute__((address_space(3)))

#if __has_builtin(__builtin_amdgcn_global_load_async_to_lds_b128) && \
    __has_builtin(__builtin_amdgcn_s_wait_asynccnt)
#define HAVE_ASYNC_LDS 1
#else
#define HAVE_ASYNC_LDS 0
#endif

// ---------------------------------------------------------------------------
// Init: out = 0, emax = -FLT_MAX, denom = 0 (every call; harness poisons ws)
// ---------------------------------------------------------------------------
__global__ void gat_init(float* __restrict__ out, float* __restrict__ emax,
                         float* __restrict__ denom) {
    int i = blockIdx.x * blockDim.x + threadIdx.x;
    if (i < GN * GHD) out[i] = 0.0f;
    if (i < GN * GH) {
        emax[i] = -3.402823466e38f;
        denom[i] = 0.0f;
    }
}

// ---------------------------------------------------------------------------
// GEMM h = feat @ W via V_WMMA_F32_16X16X4_F32.
// Block = 256 threads = 8 waves. Block tile: 32 rows x 128 cols.
// Wave w owns the 16-col strip [w*16, +16) and computes TWO 16x16 M-tiles,
// reusing each B fragment across both WMMAs.
// W (128x128, 64KB) is staged into LDS once per block — via async
// global->LDS DMA when available — with a padded stride so B-fragment
// ds_loads are bank-conflict-free.
//
// ISA layouts (wave32):
//   A 16x4 f32 : lane L -> row M=L%16 ; VGPR0=K0+(L/16)*2, VGPR1=+1
//   B 4x16 f32 : lane L -> col N=L%16 ; VGPR0=K0+(L/16)*2, VGPR1=+1
//   C/D 16x16  : lane L, vgpr v -> (M = v + 8*(L/16), N = L%16)
// ---------------------------------------------------------------------------
__global__ void gat_gemm_wmma(const float* __restrict__ feat,
                              const float* __restrict__ W,
                              float* __restrict__ h) {
    __shared__ float sW[GDIN * SWS];   // [k][col], padded stride

    const int tid  = threadIdx.x;
    const int wave = tid >> 5;
    const int lane = tid & 31;
    const int row0 = blockIdx.x * ROWS;
    const int col0 = wave * 16;
    const int mn   = lane & 15;
    const int koff = (lane >> 4) * 2;  // 0 or 2

    // ---- Stage W into LDS (16 b128 async DMA chunks per thread) ----
    {
        const float4* __restrict__ gW4 = (const float4*)W;  // 4096 chunks
        for (int idx = tid; idx < (GDIN * GHD) / 4; idx += 256) {
            int k  = idx >> 5;        // source row
            int c4 = idx & 31;        // float4 within row
#if HAVE_ASYNC_LDS
            __builtin_amdgcn_global_load_async_to_lds_b128(
                (AS1 v4i*)(gW4 + idx),
                (AS3 v4i*)(&sW[k * SWS + c4 * 4]),
                /*offset=*/0, /*cpol=*/0);
#else
            float4 v = gW4[idx];
            *(float4*)(&sW[k * SWS + c4 * 4]) = v;
#endif
        }
#if HAVE_ASYNC_LDS
        __builtin_amdgcn_s_wait_asynccnt(0);
#endif
        __syncthreads();
    }

    // ---- Compute: 2 M-tiles per wave, K = 128 in steps of 4 ----
    const float* __restrict__ a0row = feat + (size_t)(row0 + mn) * GDIN;
    const float* __restrict__ a1row = feat + (size_t)(row0 + 16 + mn) * GDIN;
    v8f c0 = {}, c1 = {};
    #pragma unroll 4
    for (int k0 = 0; k0 < GDIN; k0 += 4) {
        v2f b, a0, a1;
        b.x  = sW[(k0 + koff) * SWS + col0 + mn];
        b.y  = sW[(k0 + koff + 1) * SWS + col0 + mn];
        a0.x = a0row[k0 + koff];
        a0.y = a0row[k0 + koff + 1];
        a1.x = a1row[k0 + koff];
        a1.y = a1row[k0 + koff + 1];
        c0 = __builtin_amdgcn_wmma_f32_16x16x4_f32(
                 false, a0, false, b, (short)0, c0, false, false);
        c1 = __builtin_amdgcn_wmma_f32_16x16x4_f32(
                 false, a1, false, b, (short)0, c1, false, false);
    }

    const int mbase = (lane >> 4) * 8;
    #pragma unroll
    for (int v = 0; v < 8; ++v) {
        h[(size_t)(row0 + mbase + v) * GHD + col0 + mn]      = c0[v];
        h[(size_t)(row0 + 16 + mbase + v) * GHD + col0 + mn] = c1[v];
    }
}

// ---------------------------------------------------------------------------
// Per-(node,head) attention half-logits.
// ---------------------------------------------------------------------------
__global__ void gat_attn(const float* __restrict__ h,
                         const float* __restrict__ attn_l,
                         const float* __restrict__ attn_r,
                         float* __restrict__ el, float* __restrict__ er) {
    int i = blockIdx.x * blockDim.x + threadIdx.x;
    if (i >= GN * GH) return;
    int n = i >> 2, head = i & 3;
    const float4* __restrict__ hp = (const float4*)(h + (size_t)n * GHD + head * GD);
    const float4* __restrict__ al = (const float4*)(attn_l + head * GD);
    const float4* __restrict__ ar = (const float4*)(attn_r + head * GD);
    float sl = 0.0f, sr = 0.0f;
    #pragma unroll
    for (int q = 0; q < GD / 4; ++q) {
        float4 v = hp[q], a = al[q], b = ar[q];
        sl += v.x * a.x + v.y * a.y + v.z * a.z + v.w * a.w;
        sr += v.x * b.x + v.y * b.y + v.z * b.z + v.w * b.w;
    }
    el[i] = sl;
    er[i] = sr;
}

__device__ __forceinline__ float leaky(float x) {
    return x > 0.0f ? x : 0.2f * x;
}

// Monotonic float atomic-max via signed/unsigned integer atomics.
__device__ __forceinline__ void atomicMaxFloat(float* addr, float v) {
    if (v >= 0.0f)
        atomicMax((int*)addr, __float_as_int(v));
    else
        atomicMin((unsigned int*)addr, __float_as_uint(v));
}

// ---------------------------------------------------------------------------
// Pass 1 over edges: emax[dst,head] = max(leakyrelu(el[src]+er[dst]))
// ---------------------------------------------------------------------------
__global__ void gat_edge_max(const int* __restrict__ src,
                             const int* __restrict__ dst,
                             const float* __restrict__ el,
                             const float* __restrict__ er,
                             float* __restrict__ emax) {
    int i = blockIdx.x * blockDim.x + threadIdx.x;
    if (i >= GE * GH) return;
    int e = i >> 2, head = i & 3;
    int s = src[e], d = dst[e];
    float v = leaky(el[s * GH + head] + er[d * GH + head]);
    atomicMaxFloat(&emax[d * GH + head], v);
}

// ---------------------------------------------------------------------------
// Pass 2 over edges: denom[dst,head] += exp(logit - emax[dst,head])
// ---------------------------------------------------------------------------
__global__ void gat_edge_sum(const int* __restrict__ src,
                             const int* __restrict__ dst,
                             const float* __restrict__ el,
                             const float* __restrict__ er,
                             const float* __restrict__ emax,
                             float* __restrict__ denom) {
    int i = blockIdx.x * blockDim.x + threadIdx.x;
    if (i >= GE * GH) return;
    int e = i >> 2, head = i & 3;
    int s = src[e], d = dst[e];
    float v = leaky(el[s * GH + head] + er[d * GH + head]);
    atomicAdd(&denom[d * GH + head], __expf(v - emax[d * GH + head]));
}

// ---------------------------------------------------------------------------
// Pass 2.5: materialize softmax coefficient per (edge, head).
// ---------------------------------------------------------------------------
__global__ void gat_edge_coeff(const int* __restrict__ src,
                               const int* __restrict__ dst,
                               const float* __restrict__ el,
                               const float* __restrict__ er,
                               const float* __restrict__ emax,
                               const float* __restrict__ denom,
                               float* __restrict__ coef) {
    int i = blockIdx.x * blockDim.x + threadIdx.x;
    if (i >= GE * GH) return;
    int e = i >> 2, head = i & 3;
    int s = src[e], d = dst[e];
    float v = leaky(el[s * GH + head] + er[d * GH + head]);
    coef[i] = __expf(v - emax[d * GH + head]) / denom[d * GH + head];
}

// ---------------------------------------------------------------------------
// Pass 3 (coef path): out[dst,hd] += h[src,hd] * coef[e,head]
// ---------------------------------------------------------------------------
__global__ void gat_aggregate_coef(const int* __restrict__ src,
                                   const int* __restrict__ dst,
                                   const float* __restrict__ coef,
                                   const float* __restrict__ h,
                                   float* __restrict__ out) {
    long long i = (long long)blockIdx.x * blockDim.x + threadIdx.x;
    if (i >= (long long)GE * GHD) return;
    int e = (int)(i >> 7);
    int hd = (int)(i & 127);
    int head = hd >> 5;
    float a = coef[(size_t)e * GH + head];
    atomicAdd(&out[(size_t)dst[e] * GHD + hd],
              h[(size_t)src[e] * GHD + hd] * a);
}

// ---------------------------------------------------------------------------
// Pass 3 (fallback: recompute coefficient; used if scratch is too small)
// ---------------------------------------------------------------------------
__global__ void gat_aggregate(const int* __restrict__ src,
                              const int* __restrict__ dst,
                              const float* __restrict__ el,
                              const float* __restrict__ er,
                              const float* __restrict__ emax,
                              const float* __restrict__ denom,
                              const float* __restrict__ h,
                              float* __restrict__ out) {
    long long i = (long long)blockIdx.x * blockDim.x + threadIdx.x;
    if (i >= (long long)GE * GHD) return;
    int e = (int)(i >> 7);
    int hd = (int)(i & 127);
    int head = hd >> 5;
    int s = src[e], d = dst[e];
    float v = leaky(el[s * GH + head] + er[d * GH + head]);
    float a = __expf(v - emax[d * GH + head]) / denom[d * GH + head];
    atomicAdd(&out[(size_t)d * GHD + hd], h[(size_t)s * GHD + hd] * a);
}

// ---------------------------------------------------------------------------
// Launch
// ---------------------------------------------------------------------------
extern "C" void kernel_launch(void* const* d_in, const int* in_sizes, int n_in,
                              void* d_out, int out_size, void* d_ws, size_t ws_size,
                              hipStream_t stream) {
    const float* feat   = (const float*)d_in[0];
    const int*   src    = (const int*)d_in[1];
    const int*   dst    = (const int*)d_in[2];
    const float* W      = (const float*)d_in[3];
    const float* attn_l = (const float*)d_in[4];
    const float* attn_r = (const float*)d_in[5];
    float* out = (float*)d_out;

    // Workspace: h(N*128) | el(N*4) | er(N*4) | emax(N*4) | denom(N*4) | coef(E*4)
    float* h     = (float*)d_ws;
    float* el    = h + (size_t)GN * GHD;
    float* er    = el + (size_t)GN * GH;
    float* emax  = er + (size_t)GN * GH;
    float* denom = emax + (size_t)GN * GH;
    float* coef  = denom + (size_t)GN * GH;

    size_t base_floats = (size_t)GN * GHD + 4ull * GN * GH;
    size_t coef_floats = base_floats + (size_t)GE * GH;
    bool use_coef = ws_size >= coef_floats * sizeof(float);

    const int T = 256;

    gat_init<<<(GN * GHD + T - 1) / T, T, 0, stream>>>(out, emax, denom);

    gat_gemm_wmma<<<GN / ROWS, 256, 0, stream>>>(feat, W, h);

    gat_attn<<<(GN * GH + T - 1) / T, T, 0, stream>>>(h, attn_l, attn_r, el, er);

    gat_edge_max<<<(GE * GH + T - 1) / T, T, 0, stream>>>(src, dst, el, er, emax);

    gat_edge_sum<<<(GE * GH + T - 1) / T, T, 0, stream>>>(src, dst, el, er, emax, denom);

    long long aggN = (long long)GE * GHD;
    int aggBlocks = (int)((aggN + T - 1) / T);
    if (use_coef) {
        gat_edge_coeff<<<(GE * GH + T - 1) / T, T, 0, stream>>>(
            src, dst, el, er, emax, denom, coef);
        gat_aggregate_coef<<<aggBlocks, T, 0, stream>>>(src, dst, coef, h, out);
    } else {
        gat_aggregate<<<aggBlocks, T, 0, stream>>>(
            src, dst, el, er, emax, denom, h, out);
    }
}